// MPNN_75084618269476
// MI455X (gfx1250) — compile-verified
//
#include <hip/hip_runtime.h>
#include <hip/hip_bf16.h>

#define GN 50000
#define GE 800000
#define GIN 256
#define GH 128
#define GOUT 7
#define GEPS 1e-5f

typedef float v2f __attribute__((ext_vector_type(2)));
typedef float v8f __attribute__((ext_vector_type(8)));

// ---------------------------------------------------------------- utilities
__global__ void mpnn_zero_kernel(float* __restrict__ p, size_t n) {
    size_t i = (size_t)blockIdx.x * blockDim.x + threadIdx.x;
    size_t stride = (size_t)gridDim.x * blockDim.x;
    for (; i < n; i += stride) p[i] = 0.0f;
}

__global__ void mpnn_degree_kernel(const int* __restrict__ dst, float* __restrict__ deg, int E) {
    int i = blockIdx.x * blockDim.x + threadIdx.x;
    int stride = gridDim.x * blockDim.x;
    for (; i < E; i += stride) atomicAdd(&deg[dst[i]], 1.0f);
}

__global__ void mpnn_dinv_kernel(float* __restrict__ deg, int n) {
    int i = blockIdx.x * blockDim.x + threadIdx.x;
    int stride = gridDim.x * blockDim.x;
    for (; i < n; i += stride) deg[i] = rsqrtf(deg[i] + 1.0f);   // in-place: deg -> dinv
}

// ---------------------------------------------------------------- WMMA GEMM
// C[M,Nout] = A[M,K] @ W[K,Nout] (+bias) (+relu).
// Branch-free: requires M % 16 == 0, K % 4 == 0, Nout % (16*NT) == 0.
// One wave computes a 16 x (16*NT) strip; A fragment is reused across the
// NT column tiles. fp32 path via V_WMMA_F32_16X16X4_F32.
template <int NT>
__global__ void mpnn_gemm_wmma_kernel(const float* __restrict__ A,
                                      const float* __restrict__ W,
                                      const float* __restrict__ bias,
                                      float* __restrict__ C,
                                      int M, int K, int Nout,
                                      int doRelu) {
    const int lane = threadIdx.x & 31;
    const int wavesPerBlock = blockDim.x >> 5;
    const int wid = blockIdx.x * wavesPerBlock + (threadIdx.x >> 5);
    const int totalWaves = gridDim.x * wavesPerBlock;

    const int stripsN = Nout / (16 * NT);
    const int stripsM = M >> 4;
    const int nStrips = stripsM * stripsN;

    const int half = lane >> 4;        // 0: lanes 0-15 (K=0,1), 1: lanes 16-31 (K=2,3)
    const int l15  = lane & 15;

    for (int s = wid; s < nStrips; s += totalWaves) {
        const int tm = s / stripsN;
        const int tn = s - tm * stripsN;
        const int m0 = tm << 4;
        const int n0 = tn * (16 * NT);

        // A: lane holds row m0+l15, K-pair at (k + 2*half)        -> float2 load
        const float* Ap = A + (size_t)(m0 + l15) * K + half * 2;
        // B: lane holds col n0+l15 (+16t); rows kb, kb+1 at +0,+Nout
        const float* Bp = W + (size_t)(half * 2) * Nout + n0 + l15;

        v8f acc[NT];
        #pragma unroll
        for (int t = 0; t < NT; ++t) acc[t] = v8f{};

        for (int k = 0; k < K; k += 4) {
            v2f a;
            a.x = Ap[0];
            a.y = Ap[1];
            #pragma unroll
            for (int t = 0; t < NT; ++t) {
                v2f b;
                b.x = Bp[t * 16];          // W[kb    ][col + 16t]
                b.y = Bp[t * 16 + Nout];   // W[kb + 1][col + 16t]
                acc[t] = __builtin_amdgcn_wmma_f32_16x16x4_f32(
                    false, a, false, b, (short)0, acc[t], false, false);
            }
            Ap += 4;
            Bp += (size_t)4 * Nout;
        }

        const int crow = m0 + half * 8;    // C VGPR r -> row m0 + r + 8*half
        #pragma unroll
        for (int t = 0; t < NT; ++t) {
            const int col = n0 + t * 16 + l15;
            const float bv = bias ? bias[col] : 0.0f;
            #pragma unroll
            for (int r = 0; r < 8; ++r) {
                float v = acc[t][r] + bv;
                if (doRelu) v = fmaxf(v, 0.0f);
                C[(size_t)(crow + r) * Nout + col] = v;
            }
        }
    }
}

// ---------------------------------------------------------------- edge scatter
// One wave per edge; 128 features = 32 lanes x float4; per-float f32 atomics.
__global__ void mpnn_scatter_kernel(const int* __restrict__ src,
                                    const int* __restrict__ dst,
                                    const float* __restrict__ dinv,
                                    const float* __restrict__ hw,
                                    float* __restrict__ agg, int E) {
    const int lane = threadIdx.x & 31;
    const int wavesPerBlock = blockDim.x >> 5;
    int wid = blockIdx.x * wavesPerBlock + (threadIdx.x >> 5);
    const int totalWaves = gridDim.x * wavesPerBlock;
    for (int e = wid; e < E; e += totalWaves) {
        const int s = src[e];
        const int d = dst[e];
        const float nrm = dinv[s] * dinv[d];
        const float4 v = ((const float4*)(hw + (size_t)s * GH))[lane];
        float* ad = agg + (size_t)d * GH + lane * 4;
        atomicAdd(ad + 0, v.x * nrm);
        atomicAdd(ad + 1, v.y * nrm);
        atomicAdd(ad + 2, v.z * nrm);
        atomicAdd(ad + 3, v.w * nrm);
    }
}

// ---------------------------------------------------------------- combine
// agg <- agg + hw * dinv[row]^2 + conv_b[f] + skip  (in-place on agg)
__global__ void mpnn_combine_kernel(float* __restrict__ agg,
                                    const float* __restrict__ hw,
                                    const float* __restrict__ skip,
                                    const float* __restrict__ dinv,
                                    const float* __restrict__ convb,
                                    size_t total) {
    size_t i = (size_t)blockIdx.x * blockDim.x + threadIdx.x;
    size_t stride = (size_t)gridDim.x * blockDim.x;
    for (; i < total; i += stride) {
        const int r = (int)(i >> 7);
        const int f = (int)(i & 127);
        const float dv = dinv[r];
        agg[i] = agg[i] + hw[i] * (dv * dv) + convb[f] + skip[i];
    }
}

// ---------------------------------------------------------------- BN stats
// Per-feature sum / sumsq partials per block, then atomics into sums/sumsq.
__global__ void mpnn_bn_stats_kernel(const float* __restrict__ t,
                                     float* __restrict__ sums,
                                     float* __restrict__ sumsq,
                                     int Nrows) {
    const int f = threadIdx.x & 127;
    const int half = threadIdx.x >> 7;             // 0 or 1 (blockDim = 256)
    const int ROWS = 128;
    const int r0 = blockIdx.x * ROWS;
    const int rEnd = (r0 + ROWS < Nrows) ? (r0 + ROWS) : Nrows;
    float s = 0.0f, sq = 0.0f;
    for (int r = r0 + half; r < rEnd; r += 2) {
        const float v = t[(size_t)r * GH + f];
        s += v; sq += v * v;
    }
    __shared__ float ls[256];
    __shared__ float lq[256];
    ls[threadIdx.x] = s;
    lq[threadIdx.x] = sq;
    __syncthreads();
    if (half == 0) {
        atomicAdd(&sums[f],  s  + ls[threadIdx.x + 128]);
        atomicAdd(&sumsq[f], sq + lq[threadIdx.x + 128]);
    }
}

// ---------------------------------------------------------------- BN + LN + ReLU
// One wave per row; LN reduction over 128 features via wave32 shfl_xor.
__global__ void mpnn_bn_ln_relu_kernel(const float* __restrict__ t,
                                       const float* __restrict__ sums,
                                       const float* __restrict__ sumsq,
                                       const float* __restrict__ bng,
                                       const float* __restrict__ bnb,
                                       const float* __restrict__ lng,
                                       const float* __restrict__ lnb,
                                       float* __restrict__ out, int Nrows) {
    const int lane = threadIdx.x & 31;
    const int wavesPerBlock = blockDim.x >> 5;
    int wid = blockIdx.x * wavesPerBlock + (threadIdx.x >> 5);
    const int totalWaves = gridDim.x * wavesPerBlock;
    const float invN = 1.0f / (float)Nrows;

    for (int r = wid; r < Nrows; r += totalWaves) {
        const float4 v4 = ((const float4*)(t + (size_t)r * GH))[lane];
        float vv[4] = {v4.x, v4.y, v4.z, v4.w};
        const int f0 = lane * 4;
        float bnv[4];
        #pragma unroll
        for (int i = 0; i < 4; ++i) {
            const int f = f0 + i;
            const float mu  = sums[f] * invN;
            const float var = sumsq[f] * invN - mu * mu;
            bnv[i] = (vv[i] - mu) * rsqrtf(var + GEPS) * bng[f] + bnb[f];
        }
        float lsum = bnv[0] + bnv[1] + bnv[2] + bnv[3];
        float lsq  = bnv[0]*bnv[0] + bnv[1]*bnv[1] + bnv[2]*bnv[2] + bnv[3]*bnv[3];
        #pragma unroll
        for (int m = 16; m >= 1; m >>= 1) {
            lsum += __shfl_xor(lsum, m, 32);
            lsq  += __shfl_xor(lsq,  m, 32);
        }
        const float lmu  = lsum * (1.0f / 128.0f);
        const float lvar = lsq * (1.0f / 128.0f) - lmu * lmu;
        const float rs   = rsqrtf(lvar + GEPS);
        float4 o;
        o.x = fmaxf((bnv[0] - lmu) * rs * lng[f0 + 0] + lnb[f0 + 0], 0.0f);
        o.y = fmaxf((bnv[1] - lmu) * rs * lng[f0 + 1] + lnb[f0 + 1], 0.0f);
        o.z = fmaxf((bnv[2] - lmu) * rs * lng[f0 + 2] + lnb[f0 + 2], 0.0f);
        o.w = fmaxf((bnv[3] - lmu) * rs * lng[f0 + 3] + lnb[f0 + 3], 0.0f);
        ((float4*)(out + (size_t)r * GH))[lane] = o;
    }
}

// ---------------------------------------------------------------- elementwise add
__global__ void mpnn_add_kernel(const float* __restrict__ a, const float* __restrict__ b,
                                float* __restrict__ c, size_t n) {
    size_t i = (size_t)blockIdx.x * blockDim.x + threadIdx.x;
    size_t stride = (size_t)gridDim.x * blockDim.x;
    for (; i < n; i += stride) c[i] = a[i] + b[i];
}

// ---------------------------------------------------------------- pred padding
// wpad [128,16] = pred_W [128,7] zero-padded; bpad[16] = pred_b zero-padded.
__global__ void mpnn_pad_pred_kernel(const float* __restrict__ pw,
                                     const float* __restrict__ pb,
                                     float* __restrict__ wpad,
                                     float* __restrict__ bpad) {
    int i = blockIdx.x * blockDim.x + threadIdx.x;
    if (i < GH * 16) {
        const int r = i >> 4;
        const int c = i & 15;
        wpad[i] = (c < GOUT) ? pw[r * GOUT + c] : 0.0f;
    }
    if (i < 16) bpad[i] = (i < GOUT) ? pb[i] : 0.0f;
}

// pack padded [N,16] -> [N,7]
__global__ void mpnn_pack_out_kernel(const float* __restrict__ opad,
                                     float* __restrict__ out, int Nrows) {
    int i = blockIdx.x * blockDim.x + threadIdx.x;
    const int total = Nrows * GOUT;
    const int stride = gridDim.x * blockDim.x;
    for (; i < total; i += stride) {
        const int r = i / GOUT;
        const int c = i - r * GOUT;
        out[i] = opad[(size_t)r * 16 + c];
    }
}

// ---------------------------------------------------------------- host side
static inline void launch_gemm2(const float* A, const float* W, const float* bias,
                                float* C, int M, int K, int Nout, int relu,
                                hipStream_t stream) {
    // NT=2: 16x32 strips, Nout must be a multiple of 32
    const int strips = (M / 16) * (Nout / 32);
    int blocks = (strips + 7) / 8;                 // 8 waves / 256-thread block
    if (blocks < 1) blocks = 1;
    mpnn_gemm_wmma_kernel<2><<<blocks, 256, 0, stream>>>(A, W, bias, C, M, K, Nout, relu);
}

static inline void launch_gemm1(const float* A, const float* W, const float* bias,
                                float* C, int M, int K, int Nout, int relu,
                                hipStream_t stream) {
    // NT=1: 16x16 tiles, Nout must be a multiple of 16
    const int strips = (M / 16) * (Nout / 16);
    int blocks = (strips + 7) / 8;
    if (blocks < 1) blocks = 1;
    mpnn_gemm_wmma_kernel<1><<<blocks, 256, 0, stream>>>(A, W, bias, C, M, K, Nout, relu);
}

extern "C" void kernel_launch(void* const* d_in, const int* in_sizes, int n_in,
                              void* d_out, int out_size, void* d_ws, size_t ws_size,
                              hipStream_t stream) {
    const float* x   = (const float*)d_in[0];
    const int*   ei  = (const int*)d_in[1];
    const int*   src = ei;           // edge_index[0]
    const int*   dst = ei + GE;      // edge_index[1]

    // params: per layer i (base 2+8i): convW, convb, linW, linb, bng, bnb, lng, lnb
    const float* lin_in_W = (const float*)d_in[26];
    const float* lin_in_b = (const float*)d_in[27];
    const float* pred_W   = (const float*)d_in[28];
    const float* pred_b   = (const float*)d_in[29];

    const size_t NH = (size_t)GN * GH;
    float* ws    = (float*)d_ws;
    float* dinv  = ws;               // [N]   (degree first, then rsqrt in-place)
    float* h0    = dinv + GN;        // [N,H]
    float* hA    = h0 + NH;          // [N,H] ping
    float* hB    = hA + NH;          // [N,H] pong
    float* hw    = hB + NH;          // [N,H] (reused for h + h0)
    float* skip  = hw + NH;          // [N,H] (reused for padded pred output [N,16])
    float* agg   = skip + NH;        // [N,H] (combine writes in place)
    float* sums  = agg + NH;         // [H]
    float* sumsq = sums + GH;        // [H]
    float* wpad  = sumsq + GH;       // [128*16] padded pred_W
    float* bpad  = wpad + GH * 16;   // [16]     padded pred_b

    // ---- degrees -> dinv
    mpnn_zero_kernel<<<256, 256, 0, stream>>>(dinv, (size_t)GN);
    mpnn_degree_kernel<<<(GE + 255) / 256, 256, 0, stream>>>(dst, dinv, GE);
    mpnn_dinv_kernel<<<(GN + 255) / 256, 256, 0, stream>>>(dinv, GN);

    // ---- h0 = relu(x @ lin_in_W + b)
    launch_gemm2(x, lin_in_W, lin_in_b, h0, GN, GIN, GH, 1, stream);

    // ---- 3 GCN layers
    const float* layer_in = x;
    float* layer_out = hA;
    for (int i = 0; i < 3; ++i) {
        const int base = 2 + 8 * i;
        const float* convW = (const float*)d_in[base + 0];
        const float* convB = (const float*)d_in[base + 1];
        const float* linW  = (const float*)d_in[base + 2];
        const float* linB  = (const float*)d_in[base + 3];
        const float* bng   = (const float*)d_in[base + 4];
        const float* bnb   = (const float*)d_in[base + 5];
        const float* lng   = (const float*)d_in[base + 6];
        const float* lnb   = (const float*)d_in[base + 7];
        const int K = (i == 0) ? GIN : GH;

        // hw = h @ convW   ;   skip = h @ linW + linB
        launch_gemm2(layer_in, convW, nullptr, hw, GN, K, GH, 0, stream);
        launch_gemm2(layer_in, linW, linB, skip, GN, K, GH, 0, stream);

        // agg = scatter_add(hw[src] * norm_e) over dst
        mpnn_zero_kernel<<<4096, 256, 0, stream>>>(agg, NH);
        mpnn_scatter_kernel<<<GE / 8, 256, 0, stream>>>(src, dst, dinv, hw, agg, GE);

        // agg <- agg + hw*dinv^2 + convB + skip
        mpnn_combine_kernel<<<8192, 256, 0, stream>>>(agg, hw, skip, dinv, convB, NH);

        // BN stats then fused BN + LN + ReLU
        mpnn_zero_kernel<<<1, 256, 0, stream>>>(sums, (size_t)(2 * GH));
        mpnn_bn_stats_kernel<<<(GN + 127) / 128, 256, 0, stream>>>(agg, sums, sumsq, GN);
        mpnn_bn_ln_relu_kernel<<<(GN + 7) / 8, 256, 0, stream>>>(
            agg, sums, sumsq, bng, bnb, lng, lnb, layer_out, GN);

        layer_in = layer_out;
        layer_out = (layer_out == hA) ? hB : hA;
    }
    // after 3 layers: output of layer 2 is in hA (hA -> hB -> hA)
    const float* hFinal = hA;

    // ---- out = (h + h0) @ pred_W + pred_b  (padded to width 16, then packed)
    mpnn_add_kernel<<<8192, 256, 0, stream>>>(hFinal, h0, hw, NH);
    mpnn_pad_pred_kernel<<<(GH * 16 + 255) / 256, 256, 0, stream>>>(pred_W, pred_b, wpad, bpad);
    launch_gemm1(hw, wpad, bpad, skip /* opad [N,16] */, GN, GH, 16, 0, stream);
    mpnn_pack_out_kernel<<<(GN * GOUT + 255) / 256, 256, 0, stream>>>(skip, (float*)d_out, GN);
}